// Quanvolution2D_4801773437014
// MI455X (gfx1250) — compile-verified
//
#include <hip/hip_runtime.h>

// CDNA5 / gfx1250, wave32. One wave simulates one output pixel (3 channels).
// State (512 complex amplitudes) lives in a per-wave 4KB LDS slice.
// The fixed RX(w1) rotation layer is applied as two complex matmuls
// (16x16 x 16x32, then 16x32 x 32x32) via chained V_WMMA_F32_16X16X4_F32.

typedef __attribute__((ext_vector_type(2))) float v2f;
typedef __attribute__((ext_vector_type(8))) float v8f;

__device__ __forceinline__ void lds_fence() {
  asm volatile("s_wait_dscnt 0" ::: "memory");
}

__device__ __forceinline__ v8f wmma4(v2f a, v2f b, v8f c) {
  // D = A(16x4) * B(4x16) + C, fp32
  return __builtin_amdgcn_wmma_f32_16x16x4_f32(false, a, false, b, (short)0, c,
                                               false, false);
}

// CNOT ring: for q in 0..8: CNOT(control=q, target=(q+1)%9).
// Amplitude index: qubit i <-> bit i. 256 disjoint pairs per CNOT, 8 per lane.
__device__ __forceinline__ void cnot_ring(float* Sre, float* Sim, int lane) {
#pragma unroll
  for (int q = 0; q < 9; ++q) {
    const int qt = (q + 1) % 9;
    const int tmask = 1 << qt;
    const int cmask = 1 << q;
#pragma unroll
    for (int jj = 0; jj < 8; ++jj) {
      int k = lane + 32 * jj;                                  // pair id 0..255
      int idx0 = ((k >> qt) << (qt + 1)) | (k & (tmask - 1));  // target bit = 0
      if (idx0 & cmask) {
        int idx1 = idx0 | tmask;
        float r0 = Sre[idx0], r1 = Sre[idx1];
        Sre[idx0] = r1; Sre[idx1] = r0;
        float i0 = Sim[idx0], i1 = Sim[idx1];
        Sim[idx0] = i1; Sim[idx1] = i0;
      }
    }
    lds_fence();
  }
}

__launch_bounds__(256)
__global__ void quanv_kernel(const float* __restrict__ x,
                             const float* __restrict__ qw,
                             float* __restrict__ out) {
  constexpr int NQ = 9, B = 4, C = 3, H = 28, W = 28, HOUT = 26, WOUT = 26;
  constexpr int TOTAL = B * HOUT * WOUT;  // 2704 waves

  __shared__ float lds[8][1024];  // per-wave: [0..511]=Re, [512..1023]=Im

  const int lane = threadIdx.x & 31;
  const int warp = threadIdx.x >> 5;
  const int wid = blockIdx.x * 8 + warp;
  if (wid >= TOTAL) return;  // wave-uniform

  const int b  = wid / (HOUT * WOUT);
  const int yy = (wid / WOUT) % HOUT;
  const int xx = wid % WOUT;

  float* Sre = &lds[warp][0];
  float* Sim = &lds[warp][512];

  const int m16 = lane & 15;  // lane % 16
  const int hh  = lane >> 4;  // half-wave

  __builtin_prefetch(x + ((b * C) * H + yy) * W + xx, 0, 1);

  // ---- fixed layer-2 angles: c/s of w1_i/2 ----------------------------------
  float c1[NQ], s1[NQ];
#pragma unroll
  for (int i = 0; i < NQ; ++i) {
    float a = qw[NQ + i] * 0.5f;
    c1[i] = cosf(a);
    s1[i] = sinf(a);
  }

  // ---- G_A = RX(w1_0)x..xRX(w1_3), 16x16 complex, as WMMA A-operand ---------
  // Entry (m,k): magnitude P = prod(bit match ? cos : sin), phase (-i)^pop(m^k).
  // A layout: lane holds row m = lane%16; chunk kc, vgpr v -> K = 4kc+v+2*half.
  v2f gAr[4], gAi[4];
#pragma unroll
  for (int kc = 0; kc < 4; ++kc) {
#pragma unroll
    for (int v = 0; v < 2; ++v) {
      int k = 4 * kc + v + 2 * hh;
      int diff = m16 ^ k;
      float P = 1.0f;
#pragma unroll
      for (int i = 0; i < 4; ++i) P *= ((diff >> i) & 1) ? s1[i] : c1[i];
      int hm = __popc(diff) & 3;
      gAr[kc][v] = (hm == 0) ? P : ((hm == 2) ? -P : 0.0f);
      gAi[kc][v] = (hm == 1) ? -P : ((hm == 3) ? P : 0.0f);
    }
  }

  // ---- G_B = RX(w1_4)x..xRX(w1_8), 32x32 complex, as WMMA B-operand ---------
  // B layout: lane holds col n = lane%16 (output col c' = 16t+n);
  // chunk kc, vgpr v -> K = input col c = 4kc+v+2*half. G_B is symmetric.
  v2f gBr[2][8], gBi[2][8];
#pragma unroll
  for (int t = 0; t < 2; ++t) {
    int cp = 16 * t + m16;
#pragma unroll
    for (int kc = 0; kc < 8; ++kc) {
#pragma unroll
      for (int v = 0; v < 2; ++v) {
        int c = 4 * kc + v + 2 * hh;
        int diff = c ^ cp;
        float P = 1.0f;
#pragma unroll
        for (int i = 0; i < 5; ++i) P *= ((diff >> i) & 1) ? s1[4 + i] : c1[4 + i];
        int hm = __popc(diff) & 3;
        gBr[t][kc][v] = (hm == 0) ? P : ((hm == 2) ? -P : 0.0f);
        gBi[t][kc][v] = (hm == 1) ? -P : ((hm == 3) ? P : 0.0f);
      }
    }
  }

  float zacc[4] = {0.f, 0.f, 0.f, 0.f};

  for (int ch = 0; ch < C; ++ch) {
    // ---- merged embedding + layer-1 angles: alpha_i = p_i + w0_i ------------
    float ca[NQ], sa[NQ];
    const float* xp = x + ((b * C + ch) * H + yy) * W + xx;
#pragma unroll
    for (int dy = 0; dy < 3; ++dy) {
#pragma unroll
      for (int dx = 0; dx < 3; ++dx) {
        int i = dy * 3 + dx;
        float a = (xp[dy * W + dx] + qw[i]) * 0.5f;
        ca[i] = cosf(a);
        sa[i] = sinf(a);
      }
    }

    // ---- product state: amp(idx) = (-i)^pop(idx) * prod(bit? sin : cos) ----
#pragma unroll
    for (int j = 0; j < 16; ++j) {
      int idx = lane + 32 * j;
      float P = 1.0f;
#pragma unroll
      for (int i = 0; i < NQ; ++i) P *= ((idx >> i) & 1) ? sa[i] : ca[i];
      int hm = __popc(idx) & 3;
      Sre[idx] = (hm == 0) ? P : ((hm == 2) ? -P : 0.0f);
      Sim[idx] = (hm == 1) ? -P : ((hm == 3) ? P : 0.0f);
    }
    lds_fence();

    cnot_ring(Sre, Sim, lane);  // ring after layer-1 rotations

    // ---- left multiply: S' = G_A * S  (state as B operand) ------------------
    // State index: idx = r + 16*c, r = qubits 0-3, c = qubits 4-8.
    v2f bR[2][4], bI[2][4];
#pragma unroll
    for (int t = 0; t < 2; ++t) {
#pragma unroll
      for (int kc = 0; kc < 4; ++kc) {
#pragma unroll
        for (int v = 0; v < 2; ++v) {
          int r = 4 * kc + v + 2 * hh;
          int c = 16 * t + m16;
          bR[t][kc][v] = Sre[r + 16 * c];
          bI[t][kc][v] = Sim[r + 16 * c];
        }
      }
    }
#pragma unroll
    for (int t = 0; t < 2; ++t) {
      v8f rr = {0.f,0.f,0.f,0.f,0.f,0.f,0.f,0.f};
      v8f ii = rr, ri = rr, ir = rr;
#pragma unroll
      for (int kc = 0; kc < 4; ++kc) {
        rr = wmma4(gAr[kc], bR[t][kc], rr);  // Gr*Sr
        ii = wmma4(gAi[kc], bI[t][kc], ii);  // Gi*Si
        ri = wmma4(gAr[kc], bI[t][kc], ri);  // Gr*Si
        ir = wmma4(gAi[kc], bR[t][kc], ir);  // Gi*Sr
      }
      v8f nr = rr - ii, ni = ri + ir;
      // D layout: lane -> col c = 16t + lane%16; vgpr v -> row r = v + 8*half
#pragma unroll
      for (int v = 0; v < 8; ++v) {
        int r = v + 8 * hh;
        int c = 16 * t + m16;
        Sre[r + 16 * c] = nr[v];
        Sim[r + 16 * c] = ni[v];
      }
    }
    lds_fence();

    // ---- right multiply: S'' = S' * G_B  (state as A operand) ---------------
    v2f aR[8], aI[8];
#pragma unroll
    for (int kc = 0; kc < 8; ++kc) {
#pragma unroll
      for (int v = 0; v < 2; ++v) {
        int c = 4 * kc + v + 2 * hh;  // contracted input column
        aR[kc][v] = Sre[m16 + 16 * c];  // row m = lane%16
        aI[kc][v] = Sim[m16 + 16 * c];
      }
    }
#pragma unroll
    for (int t = 0; t < 2; ++t) {
      v8f rr = {0.f,0.f,0.f,0.f,0.f,0.f,0.f,0.f};
      v8f ii = rr, ri = rr, ir = rr;
#pragma unroll
      for (int kc = 0; kc < 8; ++kc) {
        rr = wmma4(aR[kc], gBr[t][kc], rr);  // Sr*Gr
        ii = wmma4(aI[kc], gBi[t][kc], ii);  // Si*Gi
        ri = wmma4(aI[kc], gBr[t][kc], ri);  // Si*Gr
        ir = wmma4(aR[kc], gBi[t][kc], ir);  // Sr*Gi
      }
      v8f nr = rr - ii, ni = ri + ir;
#pragma unroll
      for (int v = 0; v < 8; ++v) {
        int r = v + 8 * hh;
        int c = 16 * t + m16;
        Sre[r + 16 * c] = nr[v];
        Sim[r + 16 * c] = ni[v];
      }
    }
    lds_fence();

    cnot_ring(Sre, Sim, lane);  // ring after layer-2 rotations

    // ---- <Z_f> = sum_idx |amp|^2 * (1 - 2*bit_f(idx)), f = 0..3 -------------
    float z[4] = {0.f, 0.f, 0.f, 0.f};
#pragma unroll
    for (int j = 0; j < 16; ++j) {
      int idx = lane + 32 * j;
      float re = Sre[idx], im = Sim[idx];
      float p = re * re + im * im;
#pragma unroll
      for (int f = 0; f < 4; ++f) z[f] += ((idx >> f) & 1) ? -p : p;
    }
#pragma unroll
    for (int f = 0; f < 4; ++f) {
#pragma unroll
      for (int m = 1; m < 32; m <<= 1) z[f] += __shfl_xor(z[f], m, 32);
      zacc[f] += z[f];
    }
    lds_fence();
  }

  if (lane == 0) {
    const float inv = 1.0f / 3.0f;
#pragma unroll
    for (int f = 0; f < 4; ++f)
      out[((b * 4 + f) * HOUT + yy) * WOUT + xx] = zacc[f] * inv;
  }
}

extern "C" void kernel_launch(void* const* d_in, const int* in_sizes, int n_in,
                              void* d_out, int out_size, void* d_ws, size_t ws_size,
                              hipStream_t stream) {
  (void)in_sizes; (void)n_in; (void)out_size; (void)d_ws; (void)ws_size;
  const float* x  = (const float*)d_in[0];
  const float* qw = (const float*)d_in[1];
  float* out = (float*)d_out;
  const int total_waves = 4 * 26 * 26;  // 2704, divisible by 8
  dim3 grid(total_waves / 8), block(256);
  hipLaunchKernelGGL(quanv_kernel, grid, block, 0, stream, x, qw, out);
}